// DeepNCMDecoder_59897613910017
// MI455X (gfx1250) — compile-verified
//
#include <hip/hip_runtime.h>
#include <hip/hip_bf16.h>

typedef float v2f __attribute__((ext_vector_type(2)));
typedef float v8f __attribute__((ext_vector_type(8)));
typedef int async_v4i __attribute__((vector_size(16)));  // matches builtin's V4i

#define BM 128
#define BN 128
#define BK 16
#define LDA_S (BK + 4)   // 20 floats -> 80B row stride (16B multiple)
#define LDB_S (BN + 4)   // 132 floats -> 528B row stride (16B multiple)

// ---- CDNA5 async global->LDS path (guarded; falls back to reg pipeline) ----
#if defined(__has_builtin)
#  if __has_builtin(__builtin_amdgcn_global_load_async_to_lds_b128)
#    define GEMM_ASYNC 1
#  endif
#endif
#ifndef GEMM_ASYNC
#  define GEMM_ASYNC 0
#endif

// builtin signature (from clang diagnostic): (v4i addrspace(1)*, v4i addrspace(3)*, Imm, Imm)
#define AS1V4I(p) ((__attribute__((address_space(1))) async_v4i*)(uintptr_t)(p))
#define AS3V4I(p) ((__attribute__((address_space(3))) async_v4i*)(uint32_t)(uintptr_t)(p))

__device__ __forceinline__ void wait_async0() {
#if GEMM_ASYNC
#  if defined(__has_builtin) && __has_builtin(__builtin_amdgcn_s_wait_asynccnt)
  __builtin_amdgcn_s_wait_asynccnt(0);
#  else
  asm volatile("s_wait_asynccnt 0x0" ::: "memory");
#  endif
#endif
}

// MODE 0: C = relu(A*B + bias)          (aux0 = bias[N])
// MODE 1: C = A*B + bias                (aux0 = bias[N])
// MODE 2: C = -max(x2[m]+p2[n]-2*A*P^T, 0), B matrix = prototypes [N,K] row-major
//         (aux0 = x2[M], aux1 = p2[N]); N bounds-checked (C=1000)
template <int MODE>
__global__ __launch_bounds__(256) void gemm_wmma_f32(
    const float* __restrict__ A, const float* __restrict__ Bmat,
    const float* __restrict__ aux0, const float* __restrict__ aux1,
    float* __restrict__ Cout, int M, int N, int K, int ldb, int ldc) {
  __shared__ __align__(16) float As[2][BM * LDA_S];
  __shared__ __align__(16) float Bs[2][BK * LDB_S];

  const int tid  = threadIdx.x;
  const int lane = tid & 31;
  const int wave = tid >> 5;
  const int wm   = wave & 1;   // wave row: 0..1 (64 rows each)
  const int wn   = wave >> 1;  // wave col: 0..3 (32 cols each)
  const int m0   = blockIdx.y * BM;
  const int n0   = blockIdx.x * BN;

  const int l16 = lane & 15;
  const int lh  = lane >> 4;   // 0 or 1

  // staging index maps (256 threads)
  const int ar  = tid >> 2;        // A row 0..63 (+64)
  const int ac4 = (tid & 3) * 4;   // A col 0,4,8,12
  const int br  = tid >> 5;        // B row 0..7 (+8)           (modes 0/1)
  const int bc4 = (tid & 31) * 4;  // B col 0..124              (modes 0/1)
  const int tn  = tid & 127;       // B col 0..127              (mode 2)
  const int tq  = tid >> 7;        // B k-quad 0..1 (+2)        (mode 2)

  v8f acc[4][2];
#pragma unroll
  for (int i = 0; i < 4; ++i)
#pragma unroll
    for (int j = 0; j < 2; ++j)
#pragma unroll
      for (int v = 0; v < 8; ++v) acc[i][j][v] = 0.0f;

  float4 ra[2], rb[2];

  auto load_regs_A = [&](int kt) {
#pragma unroll
    for (int it = 0; it < 2; ++it)
      ra[it] = *reinterpret_cast<const float4*>(
          A + (size_t)(m0 + ar + it * 64) * K + kt + ac4);
  };
  auto store_regs_A = [&](int buf) {
#pragma unroll
    for (int it = 0; it < 2; ++it)
      *reinterpret_cast<float4*>(&As[buf][(ar + it * 64) * LDA_S + ac4]) = ra[it];
  };
  auto load_regs_B = [&](int kt) {
    if (MODE != 2) {
#pragma unroll
      for (int it = 0; it < 2; ++it)
        rb[it] = *reinterpret_cast<const float4*>(
            Bmat + (size_t)(kt + br + it * 8) * ldb + n0 + bc4);
    } else {
      const int col = n0 + tn;
#pragma unroll
      for (int it = 0; it < 2; ++it) {
        rb[it] = make_float4(0.f, 0.f, 0.f, 0.f);
        if (col < N)
          rb[it] = *reinterpret_cast<const float4*>(
              Bmat + (size_t)col * ldb + kt + (tq + it * 2) * 4);
      }
    }
  };
  auto store_regs_B = [&](int buf) {
    if (MODE != 2) {
#pragma unroll
      for (int it = 0; it < 2; ++it)
        *reinterpret_cast<float4*>(&Bs[buf][(br + it * 8) * LDB_S + bc4]) = rb[it];
    } else {
#pragma unroll
      for (int it = 0; it < 2; ++it) {
        const int q = tq + it * 2;  // k-quad 0..3
        Bs[buf][(q * 4 + 0) * LDB_S + tn] = rb[it].x;
        Bs[buf][(q * 4 + 1) * LDB_S + tn] = rb[it].y;
        Bs[buf][(q * 4 + 2) * LDB_S + tn] = rb[it].z;
        Bs[buf][(q * 4 + 3) * LDB_S + tn] = rb[it].w;
      }
    }
  };

#if GEMM_ASYNC
  auto stage_async = [&](int buf, int kt) {
#pragma unroll
    for (int it = 0; it < 2; ++it) {
      const float* g = A + (size_t)(m0 + ar + it * 64) * K + kt + ac4;
      const float* l = &As[buf][(ar + it * 64) * LDA_S + ac4];
      __builtin_amdgcn_global_load_async_to_lds_b128(AS1V4I(g), AS3V4I(l), 0, 0);
    }
    if (MODE != 2) {
#pragma unroll
      for (int it = 0; it < 2; ++it) {
        const float* g = Bmat + (size_t)(kt + br + it * 8) * ldb + n0 + bc4;
        const float* l = &Bs[buf][(br + it * 8) * LDB_S + bc4];
        __builtin_amdgcn_global_load_async_to_lds_b128(AS1V4I(g), AS3V4I(l), 0, 0);
      }
    }
  };
#endif

  auto compute = [&](int buf) {
#pragma unroll
    for (int kk = 0; kk < 4; ++kk) {
      const int kb = kk * 4 + lh * 2;  // this lane-half's K pair
      v2f a[4], b[2];
#pragma unroll
      for (int i = 0; i < 4; ++i) {
        const float* p = &As[buf][(wm * 64 + i * 16 + l16) * LDA_S + kb];
        a[i][0] = p[0];
        a[i][1] = p[1];
      }
#pragma unroll
      for (int j = 0; j < 2; ++j) {
        const float* p = &Bs[buf][kb * LDB_S + wn * 32 + j * 16 + l16];
        b[j][0] = p[0];
        b[j][1] = p[LDB_S];
      }
#pragma unroll
      for (int i = 0; i < 4; ++i)
#pragma unroll
        for (int j = 0; j < 2; ++j)
          acc[i][j] = __builtin_amdgcn_wmma_f32_16x16x4_f32(
              false, a[i], false, b[j], (short)0, acc[i][j], false, false);
    }
  };

  // ---- prologue: stage tile kt=0 into buffer 0 ----
#if GEMM_ASYNC
  stage_async(0, 0);
  if (MODE == 2) { load_regs_B(0); store_regs_B(0); }
  wait_async0();
#else
  load_regs_A(0);
  load_regs_B(0);
  store_regs_A(0);
  store_regs_B(0);
#endif
  __syncthreads();

  // ---- pipelined K loop: stage kt into buf^1 while computing buf ----
  int buf = 0;
  for (int kt = BK; kt < K; kt += BK) {
#if GEMM_ASYNC
    stage_async(buf ^ 1, kt);
    if (MODE == 2) load_regs_B(kt);
    compute(buf);
    if (MODE == 2) store_regs_B(buf ^ 1);
    wait_async0();
#else
    load_regs_A(kt);
    load_regs_B(kt);
    compute(buf);
    store_regs_A(buf ^ 1);
    store_regs_B(buf ^ 1);
#endif
    __syncthreads();
    buf ^= 1;
  }
  compute(buf);  // last tile

  // ---- epilogue ----
#pragma unroll
  for (int i = 0; i < 4; ++i) {
#pragma unroll
    for (int j = 0; j < 2; ++j) {
#pragma unroll
      for (int v = 0; v < 8; ++v) {
        const int m = m0 + wm * 64 + i * 16 + lh * 8 + v;
        const int n = n0 + wn * 32 + j * 16 + l16;
        float x = acc[i][j][v];
        if (MODE == 0) {
          x = fmaxf(x + aux0[n], 0.0f);
          Cout[(size_t)m * ldc + n] = x;
        } else if (MODE == 1) {
          Cout[(size_t)m * ldc + n] = x + aux0[n];
        } else {
          if (n < N) {
            const float d2 = fmaxf(aux0[m] + aux1[n] - 2.0f * x, 0.0f);
            Cout[(size_t)m * ldc + n] = -d2;
          }
        }
      }
    }
  }
}

// One wave32 per row: out[r] = sum_j X[r][j]^2
__global__ __launch_bounds__(256) void rownorm2(const float* __restrict__ X,
                                                float* __restrict__ out,
                                                int nrows, int ncols) {
  const int row  = blockIdx.x * 8 + (threadIdx.x >> 5);
  const int lane = threadIdx.x & 31;
  if (row >= nrows) return;
  const float* p = X + (size_t)row * ncols;
  float s = 0.0f;
  for (int i = lane; i < ncols; i += 32) {
    const float v = p[i];
    s += v * v;
  }
#pragma unroll
  for (int off = 16; off > 0; off >>= 1) s += __shfl_down(s, off, 32);
  if (lane == 0) out[row] = s;
}

__global__ void zero_f32(float* __restrict__ p, size_t n) {
  const size_t i = (size_t)blockIdx.x * blockDim.x + threadIdx.x;
  if (i < n) p[i] = 0.0f;
}

// one block per batch row: updates[label[b]] += enc[b]; counts[label[b]] += 1
__global__ __launch_bounds__(256) void scatter_updates(
    const float* __restrict__ enc, const int* __restrict__ labels,
    float* __restrict__ updates, float* __restrict__ counts, int E) {
  const int b = blockIdx.x;
  const int c = labels[b];
  const float* src = enc + (size_t)b * E;
  float* dst = updates + (size_t)c * E;
  for (int i = threadIdx.x; i < E; i += blockDim.x) atomicAdd(&dst[i], src[i]);
  if (threadIdx.x == 0) atomicAdd(&counts[c], 1.0f);
}

extern "C" void kernel_launch(void* const* d_in, const int* in_sizes, int n_in,
                              void* d_out, int out_size, void* d_ws,
                              size_t ws_size, hipStream_t stream) {
  const float* embedded = (const float*)d_in[0];
  const float* W1       = (const float*)d_in[1];
  const float* b1       = (const float*)d_in[2];
  const float* W2       = (const float*)d_in[3];
  const float* b2       = (const float*)d_in[4];
  const float* P        = (const float*)d_in[5];
  const int*   labels   = (const int*)d_in[6];

  const int B = 8192, D = 1024, E = 1024, C = 1000;
  const int H = 2 * E;  // 2048

  float* out     = (float*)d_out;
  float* scores  = out;                      // [B, C]
  float* updates = out + (size_t)B * C;      // [C, E]
  float* counts  = updates + (size_t)C * E;  // [C]

  float* h   = (float*)d_ws;                 // [B, H]  64 MB
  float* enc = h + (size_t)B * H;            // [B, E]  32 MB
  float* x2  = enc + (size_t)B * E;          // [B]
  float* p2  = x2 + B;                       // [C]

  {  // zero updates + counts (harness poisons d_out)
    const size_t n = (size_t)C * E + C;
    zero_f32<<<(unsigned)((n + 255) / 256), 256, 0, stream>>>(updates, n);
  }

  // h = relu(embedded @ W1 + b1)
  gemm_wmma_f32<0><<<dim3(H / BN, B / BM), 256, 0, stream>>>(
      embedded, W1, b1, nullptr, h, B, H, D, H, H);
  // enc = h @ W2 + b2
  gemm_wmma_f32<1><<<dim3(E / BN, B / BM), 256, 0, stream>>>(
      h, W2, b2, nullptr, enc, B, E, H, E, E);
  // row norms
  rownorm2<<<B / 8, 256, 0, stream>>>(enc, x2, B, E);
  rownorm2<<<(C + 7) / 8, 256, 0, stream>>>(P, p2, C, E);
  // scores = -max(x2 + p2 - 2 enc@P^T, 0)
  gemm_wmma_f32<2><<<dim3((C + BN - 1) / BN, B / BM), 256, 0, stream>>>(
      enc, P, x2, p2, scores, B, C, E, E, C);
  // prototype updates + counts
  scatter_updates<<<B, 256, 0, stream>>>(enc, labels, updates, counts, E);
}